// GCN_26276609917009
// MI455X (gfx1250) — compile-verified
//
#include <hip/hip_runtime.h>
#include <hip/hip_bf16.h>
#include <math.h>

typedef float v2f __attribute__((ext_vector_type(2)));
typedef float v8f __attribute__((ext_vector_type(8)));

#define GCN_N 50000
#define GCN_E 500000
#define GCN_F 128
#define GCN_FO 16

// ---------------- degree / normalization ----------------

__global__ void deg_init_kernel(float* __restrict__ deg, int n) {
    int i = blockIdx.x * blockDim.x + threadIdx.x;
    if (i < n) deg[i] = 1.0f;   // self-loop weight 1
}

__global__ void deg_accum_kernel(float* __restrict__ deg, const int* __restrict__ dst,
                                 const float* __restrict__ ew, int e) {
    int i = blockIdx.x * blockDim.x + threadIdx.x;
    if (i < e) unsafeAtomicAdd(&deg[dst[i]], ew[i]);
}

__global__ void deg_finalize_kernel(float* __restrict__ deg, int n) {
    int i = blockIdx.x * blockDim.x + threadIdx.x;
    if (i < n) deg[i] = rsqrtf(deg[i]);   // deg >= 1 always
}

// ---------------- GEMM: C[n x fout] = relu?(A)[n x fin] * W[fin x fout] ----------------
// One wave32 computes one 16x16 output tile via V_WMMA_F32_16X16X4_F32.
// A-frag (16x4 f32): lanes 0-15 -> K=k0,k0+1 ; lanes 16-31 -> K=k0+2,k0+3 (M = lane&15)
// B-frag (4x16 f32): v0 = W[k0+2*hi][col0+lo], v1 = W[k0+2*hi+1][col0+lo]
// C/D  (16x16 f32): VGPR j -> M = j + 8*hi, N = col0 + lo
__global__ void __launch_bounds__(256)
gemm_wmma_kernel(const float* __restrict__ A, const float* __restrict__ W,
                 float* __restrict__ C, int n, int fin, int fout, int relu) {
    const int wave = threadIdx.x >> 5;
    const int lane = threadIdx.x & 31;
    const int nTiles = n >> 4;                 // n divisible by 16
    const int rowTile = blockIdx.x * 8 + wave; // 8 waves per block
    if (rowTile >= nTiles) return;             // wave-uniform exit: EXEC stays all-1s
    const int col0 = blockIdx.y << 4;
    const int hi = lane >> 4;                  // 0 or 1
    const int lo = lane & 15;
    const int row = (rowTile << 4) + lo;

    const float* __restrict__ ap = A + (size_t)row * fin + 2 * hi;
    const float* __restrict__ bp = W + (size_t)(2 * hi) * fout + col0 + lo;

    v8f acc = {0.f, 0.f, 0.f, 0.f, 0.f, 0.f, 0.f, 0.f};
    for (int k0 = 0; k0 < fin; k0 += 4) {
        float a0 = ap[k0 + 0];
        float a1 = ap[k0 + 1];
        if (relu) { a0 = fmaxf(a0, 0.f); a1 = fmaxf(a1, 0.f); }
        v2f a = {a0, a1};
        v2f b;
        b.x = bp[(size_t)k0 * fout];
        b.y = bp[(size_t)(k0 + 1) * fout];
        acc = __builtin_amdgcn_wmma_f32_16x16x4_f32(
            /*neg_a=*/false, a, /*neg_b=*/false, b,
            /*c_mod=*/(short)0, acc, /*reuse_a=*/false, /*reuse_b=*/false);
    }

    float* __restrict__ cp = C + (size_t)((rowTile << 4) + 8 * hi) * fout + col0 + lo;
#pragma unroll
    for (int j = 0; j < 8; ++j)
        cp[(size_t)j * fout] = acc[j];
}

// ---------------- per-node init: out = xw * dinv^2 + bias ----------------

__global__ void node_init_kernel(float* __restrict__ out, const float* __restrict__ xw,
                                 const float* __restrict__ dinv, const float* __restrict__ bias,
                                 int n, int f) {
    int gid = blockIdx.x * blockDim.x + threadIdx.x;
    if (gid >= n * f) return;
    int i = gid / f;
    int c = gid - i * f;
    float di = dinv[i];
    out[gid] = xw[gid] * (di * di) + bias[c];
}

// ---------------- edge scatter-add: out[dst] += xw[src] * norm ----------------
// F=128: one wave32 per edge, 4 features per lane (coalesced 128B reads of xw[src]).
__global__ void edge_scatter128_kernel(float* __restrict__ out, const float* __restrict__ xw,
                                       const int* __restrict__ src, const int* __restrict__ dst,
                                       const float* __restrict__ ew, const float* __restrict__ dinv,
                                       int e) {
    int gid = blockIdx.x * blockDim.x + threadIdx.x;
    int ei = gid >> 5;
    if (ei >= e) return;
    int lane = gid & 31;
    int s = src[ei], d = dst[ei];
    float norm = dinv[s] * ew[ei] * dinv[d];
    const float* __restrict__ xp = xw + (size_t)s * 128 + lane;
    float* __restrict__ op = out + (size_t)d * 128 + lane;
#pragma unroll
    for (int j = 0; j < 4; ++j)
        unsafeAtomicAdd(op + 32 * j, xp[32 * j] * norm);
}

// F=16: 16 lanes per edge.
__global__ void edge_scatter16_kernel(float* __restrict__ out, const float* __restrict__ xw,
                                      const int* __restrict__ src, const int* __restrict__ dst,
                                      const float* __restrict__ ew, const float* __restrict__ dinv,
                                      int e) {
    int gid = blockIdx.x * blockDim.x + threadIdx.x;
    int ei = gid >> 4;
    if (ei >= e) return;
    int lane = gid & 15;
    int s = src[ei], d = dst[ei];
    float norm = dinv[s] * ew[ei] * dinv[d];
    unsafeAtomicAdd(out + (size_t)d * 16 + lane, xw[(size_t)s * 16 + lane] * norm);
}

// ---------------- in-place log-softmax over 16 columns ----------------

__global__ void logsoftmax16_kernel(float* __restrict__ out, int n) {
    int i = blockIdx.x * blockDim.x + threadIdx.x;
    if (i >= n) return;
    float* p = out + (size_t)i * 16;
    float v[16];
    float m = -INFINITY;
#pragma unroll
    for (int j = 0; j < 16; ++j) { v[j] = p[j]; m = fmaxf(m, v[j]); }
    float s = 0.f;
#pragma unroll
    for (int j = 0; j < 16; ++j) s += expf(v[j] - m);
    float ls = logf(s) + m;
#pragma unroll
    for (int j = 0; j < 16; ++j) p[j] = v[j] - ls;
}

// ---------------- driver ----------------

extern "C" void kernel_launch(void* const* d_in, const int* in_sizes, int n_in,
                              void* d_out, int out_size, void* d_ws, size_t ws_size,
                              hipStream_t stream) {
    const float* x   = (const float*)d_in[0];
    const int*   es  = (const int*)d_in[1];
    const int*   ed  = (const int*)d_in[2];
    const float* ew  = (const float*)d_in[3];
    const float* W1  = (const float*)d_in[4];
    const float* b1  = (const float*)d_in[5];
    const float* W2  = (const float*)d_in[6];
    const float* b2  = (const float*)d_in[7];
    const float* W3  = (const float*)d_in[8];
    const float* b3  = (const float*)d_in[9];
    float* out = (float*)d_out;

    const int N = GCN_N, E = GCN_E, F = GCN_F, FO = GCN_FO;

    // Workspace layout: dinv | xw | h  (~51.5 MB total)
    float* dinv = (float*)d_ws;
    float* xw   = dinv + 50048;              // keep 256B alignment
    float* h    = xw + (size_t)N * F;

    dim3 blk(256);

    // --- degree normalization: dinv = (1 + sum_in ew)^-1/2 ---
    deg_init_kernel<<<(N + 255) / 256, blk, 0, stream>>>(dinv, N);
    deg_accum_kernel<<<(E + 255) / 256, blk, 0, stream>>>(dinv, ed, ew, E);
    deg_finalize_kernel<<<(N + 255) / 256, blk, 0, stream>>>(dinv, N);

    const int rowBlocks = ((N >> 4) + 7) / 8;           // 391
    const int nf  = N * F;                              // 6.4M
    const int nfo = N * FO;                             // 800K

    // ---------- Layer 1: xw = x @ W1 ; h = agg + selfloop + b1 ----------
    gemm_wmma_kernel<<<dim3(rowBlocks, F / 16), blk, 0, stream>>>(x, W1, xw, N, F, F, 0);
    node_init_kernel<<<(nf + 255) / 256, blk, 0, stream>>>(h, xw, dinv, b1, N, F);
    edge_scatter128_kernel<<<(E * 32 + 255) / 256, blk, 0, stream>>>(h, xw, es, ed, ew, dinv, E);

    // ---------- Layer 2: xw = relu(h) @ W2 ; h = agg + selfloop + b2 ----------
    gemm_wmma_kernel<<<dim3(rowBlocks, F / 16), blk, 0, stream>>>(h, W2, xw, N, F, F, 1);
    node_init_kernel<<<(nf + 255) / 256, blk, 0, stream>>>(h, xw, dinv, b2, N, F);
    edge_scatter128_kernel<<<(E * 32 + 255) / 256, blk, 0, stream>>>(h, xw, es, ed, ew, dinv, E);

    // ---------- Layer 3: xw16 = relu(h) @ W3 ; out = agg + selfloop + b3 ----------
    gemm_wmma_kernel<<<dim3(rowBlocks, FO / 16), blk, 0, stream>>>(h, W3, xw, N, F, FO, 1);
    node_init_kernel<<<(nfo + 255) / 256, blk, 0, stream>>>(out, xw, dinv, b3, N, FO);
    edge_scatter16_kernel<<<(E * 16 + 255) / 256, blk, 0, stream>>>(out, xw, es, ed, ew, dinv, E);

    // ---------- log_softmax over the 16 output columns ----------
    logsoftmax16_kernel<<<(N + 255) / 256, blk, 0, stream>>>(out, N);
}